// CascadedCodebook_36816459661785
// MI455X (gfx1250) — compile-verified
//
#include <hip/hip_runtime.h>
#include <stdint.h>

// CascadedCodebook gather for MI455X (gfx1250).
// Memory-bound: 512 MB of write-once output; table (512 KB) stays L2-resident.
// Strategy: async global->LDS staging of indices (ASYNCcnt path), B128 table
// loads, non-temporal B128 output stores.

typedef float v4f __attribute__((ext_vector_type(4)));
typedef __attribute__((address_space(1))) int  gint_t;  // global
typedef __attribute__((address_space(3))) int  lint_t;  // LDS

#define EMBED_DIM     512
#define V4_PER_ROW    (EMBED_DIM / 4)   // 128 float4 per row
#define TIER0_N       16
#define TIER01_N      128               // TIER0 + TIER1
#define TOTAL_N       256
#define ROWS_PER_BLOCK 128
#define THREADS       256
#define WAVES         (THREADS / 32)

#if __has_builtin(__builtin_amdgcn_global_load_async_to_lds_b32)
#define HAVE_ASYNC_LDS 1
#endif

__global__ __launch_bounds__(THREADS)
void CascadedCodebook_gather_kernel(const int* __restrict__ idxs,
                                    const float* __restrict__ t0,
                                    const float* __restrict__ t1,
                                    const float* __restrict__ t2,
                                    float* __restrict__ out,
                                    int batch)
{
    __shared__ int s_idx[ROWS_PER_BLOCK];

    const int tid  = threadIdx.x;
    const int lane = tid & 31;
    const int wave = tid >> 5;
    const long long row0 = (long long)blockIdx.x * ROWS_PER_BLOCK;

    // ---- Stage this block's indices into LDS --------------------------------
#ifdef HAVE_ASYNC_LDS
    // CDNA5 async global->LDS copy; tracked by ASYNCcnt.
    if (tid < ROWS_PER_BLOCK && (row0 + tid) < batch) {
        gint_t* gsrc = (gint_t*)(uintptr_t)(const void*)(idxs + row0 + tid);
        lint_t* ldst = (lint_t*)(uint32_t)(uintptr_t)(void*)(s_idx + tid);
        __builtin_amdgcn_global_load_async_to_lds_b32(gsrc, ldst,
                                                      /*offset=*/0, /*cpol=*/0);
    }
#if __has_builtin(__builtin_amdgcn_s_wait_asynccnt)
    __builtin_amdgcn_s_wait_asynccnt(0);
#else
    asm volatile("s_wait_asynccnt 0" ::: "memory");
#endif
#else
    if (tid < ROWS_PER_BLOCK && (row0 + tid) < batch) {
        s_idx[tid] = idxs[row0 + tid];
    }
#endif
    __syncthreads();

    // ---- Gather: one wave per row, 4x float4 per lane -----------------------
    for (int r = wave; r < ROWS_PER_BLOCK; r += WAVES) {
        const long long row = row0 + r;
        if (row >= batch) break;

        const int idx = s_idx[r];          // wave-uniform LDS broadcast
        v4f v0, v1, v2, v3;

        if ((unsigned)idx < (unsigned)TOTAL_N) {
            // Tier select is wave-uniform -> uniform branches, scalar-friendly.
            const float* src;
            if (idx < TIER0_N)        src = t0 + (size_t)idx * EMBED_DIM;
            else if (idx < TIER01_N)  src = t1 + (size_t)(idx - TIER0_N) * EMBED_DIM;
            else                      src = t2 + (size_t)(idx - TIER01_N) * EMBED_DIM;
            const v4f* s4 = (const v4f*)src;        // L2-hot, cache normally
            v0 = s4[lane +  0];
            v1 = s4[lane + 32];
            v2 = s4[lane + 64];
            v3 = s4[lane + 96];
        } else {
            v0 = 0.0f; v1 = 0.0f; v2 = 0.0f; v3 = 0.0f;
        }

        // Streaming 512 MB write-once output: non-temporal B128 stores.
        v4f* o4 = (v4f*)(out + row * (long long)EMBED_DIM);
        __builtin_nontemporal_store(v0, o4 + lane +  0);
        __builtin_nontemporal_store(v1, o4 + lane + 32);
        __builtin_nontemporal_store(v2, o4 + lane + 64);
        __builtin_nontemporal_store(v3, o4 + lane + 96);
    }
}

extern "C" void kernel_launch(void* const* d_in, const int* in_sizes, int n_in,
                              void* d_out, int out_size, void* d_ws, size_t ws_size,
                              hipStream_t stream) {
    (void)n_in; (void)out_size; (void)d_ws; (void)ws_size;
    const int*   idxs = (const int*)d_in[0];
    const float* t0   = (const float*)d_in[1];
    const float* t1   = (const float*)d_in[2];
    const float* t2   = (const float*)d_in[3];
    float*       out  = (float*)d_out;

    const int batch  = in_sizes[0];
    const int blocks = (batch + ROWS_PER_BLOCK - 1) / ROWS_PER_BLOCK;
    CascadedCodebook_gather_kernel<<<blocks, THREADS, 0, stream>>>(
        idxs, t0, t1, t2, out, batch);
}